// AttractorLayerUnnormed_16312285790415
// MI455X (gfx1250) — compile-verified
//
#include <hip/hip_runtime.h>
#include <hip/hip_bf16.h>

typedef __attribute__((ext_vector_type(2))) float v2f;
typedef __attribute__((ext_vector_type(8))) float v8f;

#define ALPHA 300.0f

// ---------------------------------------------------------------------------
// Kernel 0: pack w1 [128 o][256 k] into WMMA-A-operand order in workspace.
// A-matrix layout for V_WMMA_F32_16X16X4_F32 (16x4, f32):
//   lane l (<16):  V0 = A[M=l, K=0], V1 = A[M=l, K=1]
//   lane l+16:     V0 = A[M=l, K=2], V1 = A[M=l, K=3]
// Packed as float2 array: w1P2[(kstep*2 + hi)*128 + o] = {w1[o][k], w1[o][k+1]}
// ---------------------------------------------------------------------------
__global__ __launch_bounds__(256) void prepack_w1_kernel(
    const float* __restrict__ w1, float* __restrict__ w1P) {
  int i = blockIdx.x * 256 + threadIdx.x;   // 128*256 = 32768 elements
  if (i >= 128 * 256) return;
  int o = i >> 8;          // 0..127
  int k = i & 255;         // 0..255
  int idx = ((((k >> 2) << 1) + ((k >> 1) & 1)) * 128 + o) * 2 + (k & 1);
  w1P[idx] = w1[i];
}

// ---------------------------------------------------------------------------
// Fused kernel: GEMM1(WMMA f32) + bias + ReLU -> LDS transpose ->
//               GEMM2(WMMA f32) + bias + softplus -> bilinear + attractor.
// Block = 128 threads = 4 waves; each wave owns 32 consecutive pixels
// (never crossing a row since W=128 is a multiple of 32).
// ---------------------------------------------------------------------------
__global__ __launch_bounds__(128) void attractor_fused_kernel(
    const float* __restrict__ x,      // [4,256,96,128]
    const float* __restrict__ bprev,  // [4,64,48,64]
    const float* __restrict__ w1P,    // packed A operands (workspace)
    const float* __restrict__ b1,     // [128]
    const float* __restrict__ w2,     // [16,128]
    const float* __restrict__ b2,     // [16]
    float* __restrict__ out)          // [4,64,96,128]
{
  __shared__ float s_w2P[2048];          // w2 packed into A-operand order
  __shared__ float s_b1[128];
  __shared__ float s_b2[16];
  __shared__ float s_hid[4][128 * 33];   // per-wave hid transpose, stride 33

  const int tid = threadIdx.x;

  // ---- build packed w2 + biases in LDS ----
  for (int i = tid; i < 2048; i += 128) {
    int k = i >> 4;        // 0..127
    int a = i & 15;        // 0..15
    int idx = ((((k >> 2) << 1) + ((k >> 1) & 1)) * 16 + a) * 2 + (k & 1);
    s_w2P[idx] = w2[a * 128 + k];
  }
  if (tid < 128) s_b1[tid] = b1[tid];
  if (tid < 16)  s_b2[tid] = b2[tid];
  __syncthreads();

  const int lane = tid & 31;
  const int widx = tid >> 5;
  const int lm   = lane & 15;
  const int hi   = lane >> 4;            // 0 or 1
  const int koff = hi << 1;              // K offset for B operand halves

  const int gwid = blockIdx.x * 4 + widx;     // 0..1535
  const int p0   = gwid * 32;                 // first pixel of this wave
  const int n    = p0 / 12288;                // batch
  const int rem  = p0 - n * 12288;
  const int h    = rem >> 7;                  // row (0..95)
  const int w0   = rem & 127;                 // 0,32,64,96

  // ---------------- GEMM1: hid[128 x 32px] = W1 * x ----------------
  v8f acc[8][2];
  {
    v8f z = {};
#pragma unroll
    for (int t = 0; t < 8; ++t)
#pragma unroll
      for (int s = 0; s < 2; ++s) acc[t][s] = z;
  }

  const float* xbase = x + (size_t)n * 256 * 12288 + h * 128 + w0;
  const v2f*   w1P2  = (const v2f*)w1P;

  for (int kk = 0; kk < 64; ++kk) {
    // A operands: 8 M-tiles of w1, one coalesced b64 load per tile
    v2f a[8];
#pragma unroll
    for (int t = 0; t < 8; ++t)
      a[t] = w1P2[(kk * 2 + hi) * 128 + t * 16 + lm];

    // B operands: x tile, K rows kk*4 + {0,1}|{2,3}
    v2f bb[2];
#pragma unroll
    for (int s = 0; s < 2; ++s) {
      const float* xp = xbase + (size_t)(kk * 4 + koff) * 12288 + s * 16 + lm;
      bb[s].x = xp[0];
      bb[s].y = xp[12288];
    }
    if (kk + 1 < 64)
      __builtin_prefetch(xbase + (size_t)((kk + 1) * 4 + koff) * 12288 + lm, 0, 0);

#pragma unroll
    for (int t = 0; t < 8; ++t)
#pragma unroll
      for (int s = 0; s < 2; ++s)
        acc[t][s] = __builtin_amdgcn_wmma_f32_16x16x4_f32(
            false, a[t], false, bb[s], (short)0, acc[t][s], false, false);
  }

  // ---- bias + ReLU, D-layout -> LDS (row stride 33 floats) ----
  float* hw = &s_hid[widx][0];
#pragma unroll
  for (int t = 0; t < 8; ++t) {
#pragma unroll
    for (int r = 0; r < 8; ++r) {
      const int o  = t * 16 + hi * 8 + r;
      const float bv = s_b1[o];
#pragma unroll
      for (int s = 0; s < 2; ++s) {
        float hval = acc[t][s][r] + bv;
        hval = fmaxf(hval, 0.0f);
        hw[o * 33 + s * 16 + lm] = hval;
      }
    }
  }
  __syncthreads();

  // ---------------- GEMM2: A[16 x 32px] = W2 * hid ----------------
  v8f a2[2];
  {
    v8f z = {};
    a2[0] = z; a2[1] = z;
  }
  const v2f* w2P2 = (const v2f*)s_w2P;
  for (int kk = 0; kk < 32; ++kk) {
    v2f wa = w2P2[(kk * 2 + hi) * 16 + lm];
#pragma unroll
    for (int s = 0; s < 2; ++s) {
      v2f bb;
      bb.x = hw[(kk * 4 + koff)     * 33 + s * 16 + lm];
      bb.y = hw[(kk * 4 + koff + 1) * 33 + s * 16 + lm];
      a2[s] = __builtin_amdgcn_wmma_f32_16x16x4_f32(
          false, wa, false, bb, (short)0, a2[s], false, false);
    }
  }

  // ---- bias + softplus (stable form: max(v,0)+log1p(exp(-|v|))) ----
#pragma unroll
  for (int s = 0; s < 2; ++s)
#pragma unroll
    for (int r = 0; r < 8; ++r) {
      float v = a2[s][r] + s_b2[hi * 8 + r];
      a2[s][r] = fmaxf(v, 0.0f) + __logf(1.0f + __expf(-fabsf(v)));
    }

  // ---------------- Stage 2: bilinear centers + exp attractor ----------------
  const float* bpn = bprev + (size_t)n * 64 * 48 * 64;

  const float ysf = (float)h * (47.0f / 95.0f);
  const int   y0  = (int)ysf;
  const int   y1  = (y0 + 1 < 47) ? (y0 + 1) : 47;
  const float wy  = ysf - (float)y0;

  int   x0v[2], x1v[2];
  float wxv[2];
#pragma unroll
  for (int s = 0; s < 2; ++s) {
    const int wpix = w0 + s * 16 + lm;
    const float xsf = (float)wpix * (63.0f / 127.0f);
    const int x0 = (int)xsf;
    x0v[s] = x0;
    x1v[s] = (x0 + 1 < 63) ? (x0 + 1) : 63;
    wxv[s] = xsf - (float)x0;
  }

  for (int j = 0; j < 64; ++j) {
    const float* r0 = bpn + (j * 48 + y0) * 64;
    const float* r1 = bpn + (j * 48 + y1) * 64;
#pragma unroll
    for (int s = 0; s < 2; ++s) {
      const float v00 = r0[x0v[s]], v01 = r0[x1v[s]];
      const float v10 = r1[x0v[s]], v11 = r1[x1v[s]];
      const float top = v00 + (v01 - v00) * wxv[s];
      const float bot = v10 + (v11 - v10) * wxv[s];
      const float bc  = top + (bot - top) * wy;   // bilinear center

      float sum = 0.0f;
#pragma unroll
      for (int r = 0; r < 8; ++r) {
        const float dx = a2[s][r] - bc;
        sum += __expf(-ALPHA * dx * dx) * dx;     // exp attractor
      }
      // combine the two half-waves (attrs 0-7 in lanes 0-15, 8-15 in 16-31)
      sum += __shfl_xor(sum, 16, 32);

      if (hi == 0) {
        out[(((size_t)n * 64 + j) * 96 + h) * 128 + (w0 + s * 16 + lm)] = bc + sum;
      }
    }
  }
}

extern "C" void kernel_launch(void* const* d_in, const int* in_sizes, int n_in,
                              void* d_out, int out_size, void* d_ws, size_t ws_size,
                              hipStream_t stream) {
  const float* x     = (const float*)d_in[0];
  const float* bprev = (const float*)d_in[1];
  const float* w1    = (const float*)d_in[2];
  const float* b1    = (const float*)d_in[3];
  const float* w2    = (const float*)d_in[4];
  const float* b2    = (const float*)d_in[5];
  float* out = (float*)d_out;
  float* w1P = (float*)d_ws;   // 128*256 floats = 128 KB packed A operands

  prepack_w1_kernel<<<128, 256, 0, stream>>>(w1, w1P);

  // 49152 pixels / 32 per wave = 1536 waves; 4 waves per block -> 384 blocks
  attractor_fused_kernel<<<384, 128, 0, stream>>>(x, bprev, w1P, b1, w2, b2, out);

  (void)in_sizes; (void)n_in; (void)out_size; (void)ws_size;
}